// TransformerWithSelfAttention_23364622090387
// MI455X (gfx1250) — compile-verified
//
#include <hip/hip_runtime.h>
#include <math.h>

#define C_DIM 128
#define N_DIM 4096
#define FF_DIM 2048
#define EPS 1e-5f

typedef float v2f __attribute__((ext_vector_type(2)));
typedef float v8f __attribute__((ext_vector_type(8)));

// D = A(16x4, f32) * B(4x16, f32) + C(16x16, f32)  -> v_wmma_f32_16x16x4_f32
__device__ __forceinline__ v8f wmma_f32(v2f a, v2f b, v8f c) {
  return __builtin_amdgcn_wmma_f32_16x16x4_f32(
      /*neg_a=*/false, a, /*neg_b=*/false, b,
      /*c_mod=*/(short)0, c, /*reuse_a=*/false, /*reuse_b=*/false);
}

// Row reductions across the 16 lanes that hold one matrix row (wave32).
__device__ __forceinline__ float redmax16(float t) {
  t = fmaxf(t, __shfl_xor(t, 8, 16));
  t = fmaxf(t, __shfl_xor(t, 4, 16));
  t = fmaxf(t, __shfl_xor(t, 2, 16));
  t = fmaxf(t, __shfl_xor(t, 1, 16));
  return t;
}
__device__ __forceinline__ float redsum16(float t) {
  t += __shfl_xor(t, 8, 16);
  t += __shfl_xor(t, 4, 16);
  t += __shfl_xor(t, 2, 16);
  t += __shfl_xor(t, 1, 16);
  return t;
}

// ---------------------------------------------------------------------------
// Kernel 1: Q/K/V projections. out[o,n] = sum_c W[o,c] x[c,n] + b[o]
// grid = (256 ntiles, 8 mtiles, 3 which), block = 32 (one wave per 16x16 tile)
// ---------------------------------------------------------------------------
__global__ void qkv_proj_kernel(const float* __restrict__ x,
                                const float* __restrict__ Wq, const float* __restrict__ bq,
                                const float* __restrict__ Wk, const float* __restrict__ bk,
                                const float* __restrict__ Wv, const float* __restrict__ bv,
                                float* __restrict__ qo, float* __restrict__ ko,
                                float* __restrict__ vo) {
  const int lane = threadIdx.x & 31;
  const int l16 = lane & 15;
  const int hw = lane >> 4;
  const int koff = hw * 2;
  const int n0 = blockIdx.x * 16;
  const int m0 = blockIdx.y * 16;
  const int which = blockIdx.z;

  const float* W = (which == 0) ? Wq : ((which == 1) ? Wk : Wv);
  const float* bias = (which == 0) ? bq : ((which == 1) ? bk : bv);
  float* out = (which == 0) ? qo : ((which == 1) ? ko : vo);

  v8f acc = {};
#pragma unroll 8
  for (int k0 = 0; k0 < C_DIM; k0 += 4) {
    v2f a, b;
    a.x = W[(m0 + l16) * C_DIM + k0 + koff];
    a.y = W[(m0 + l16) * C_DIM + k0 + koff + 1];
    b.x = x[(size_t)(k0 + koff) * N_DIM + n0 + l16];
    b.y = x[(size_t)(k0 + koff + 1) * N_DIM + n0 + l16];
    acc = wmma_f32(a, b, acc);
  }
#pragma unroll
  for (int r = 0; r < 8; ++r) {
    const int o = m0 + r + 8 * hw;
    out[(size_t)o * N_DIM + n0 + l16] = acc[r] + bias[o];
  }
}

// ---------------------------------------------------------------------------
// Kernel 2: flash attention per (head, 16-query block). One wave per block.
// scores[i,j] = 0.25 * sum_c q[c,i] k[c,j]; online softmax; acc[i,c] += P·Vt
// q/k/v stored [C][N]; head h occupies rows 16h..16h+15.
// ---------------------------------------------------------------------------
__global__ void attention_kernel(const float* __restrict__ q,
                                 const float* __restrict__ k,
                                 const float* __restrict__ v,
                                 float* __restrict__ attn_out) {
  __shared__ float ptile[16 * 16];
  const int lane = threadIdx.x & 31;
  const int l16 = lane & 15;
  const int hw = lane >> 4;
  const int koff = hw * 2;
  const int qb = blockIdx.x;   // 0..255
  const int h = blockIdx.y;    // 0..7

  const float* qg = q + (size_t)h * 16 * N_DIM;
  const float* kg = k + (size_t)h * 16 * N_DIM;
  const float* vg = v + (size_t)h * 16 * N_DIM;

  // Q tile as A fragments (A[i,c] = q[c, qb*16+i]), loaded once.
  v2f aq[4];
#pragma unroll
  for (int s = 0; s < 4; ++s) {
    aq[s].x = qg[(size_t)(s * 4 + koff) * N_DIM + qb * 16 + l16];
    aq[s].y = qg[(size_t)(s * 4 + koff + 1) * N_DIM + qb * 16 + l16];
  }

  v8f accA = {};                       // [i, c] accumulator
  float m_run[8], l_run[8];
#pragma unroll
  for (int r = 0; r < 8; ++r) { m_run[r] = -1.0e30f; l_run[r] = 0.0f; }

  for (int jb = 0; jb < N_DIM / 16; ++jb) {
    // Prefetch next key/value tiles (global_prefetch_b8).
    if (jb + 1 < N_DIM / 16) {
      __builtin_prefetch(&kg[(size_t)koff * N_DIM + (jb + 1) * 16 + l16], 0, 3);
      __builtin_prefetch(&vg[(size_t)l16 * N_DIM + (jb + 1) * 16 + koff], 0, 3);
    }
    // Scores tile: S[i,j] = sum_c q[c,i] k[c,j]
    v8f S = {};
#pragma unroll
    for (int s = 0; s < 4; ++s) {
      v2f bk;
      bk.x = kg[(size_t)(s * 4 + koff) * N_DIM + jb * 16 + l16];
      bk.y = kg[(size_t)(s * 4 + koff + 1) * N_DIM + jb * 16 + l16];
      S = wmma_f32(aq[s], bk, S);
    }
    // Online softmax update; stage P to LDS (D-layout -> [i][j]).
#pragma unroll
    for (int r = 0; r < 8; ++r) {
      const float s = S[r] * 0.25f;            // 1/sqrt(hd), hd=16
      const float rm = redmax16(s);
      const float mo = m_run[r];
      const float mn = fmaxf(mo, rm);
      const float al = __expf(mo - mn);
      const float p = __expf(s - mn);
      l_run[r] = l_run[r] * al + redsum16(p);
      m_run[r] = mn;
      accA[r] *= al;
      ptile[(r + 8 * hw) * 16 + l16] = p;
    }
    __syncthreads();
    // acc[i,c] += P[i,j] * v[c,j]   (A = P from LDS, B[j,c] = v[c, jb*16+j])
#pragma unroll
    for (int s = 0; s < 4; ++s) {
      v2f ap, bv;
      ap.x = ptile[l16 * 16 + s * 4 + koff];
      ap.y = ptile[l16 * 16 + s * 4 + koff + 1];
      bv.x = vg[(size_t)l16 * N_DIM + jb * 16 + s * 4 + koff];
      bv.y = vg[(size_t)l16 * N_DIM + jb * 16 + s * 4 + koff + 1];
      accA = wmma_f32(ap, bv, accA);
    }
    __syncthreads();
  }
  // Write attn as [C][N]: row h*16 + c, col qb*16 + i  (element (i=r+8hw, c=l16))
#pragma unroll
  for (int r = 0; r < 8; ++r) {
    attn_out[(size_t)(h * 16 + l16) * N_DIM + qb * 16 + r + 8 * hw] =
        accA[r] / l_run[r];
  }
}

// ---------------------------------------------------------------------------
// Kernel 3: Wo projection + residual + LayerNorm1.
// One wave per 16-voxel column; computes the whole 128-channel column.
// Writes x1 as [N][C] for the FFN kernel.
// ---------------------------------------------------------------------------
__global__ void proj_ln1_kernel(const float* __restrict__ attn,
                                const float* __restrict__ Wo,
                                const float* __restrict__ bo,
                                const float* __restrict__ x,
                                const float* __restrict__ g1,
                                const float* __restrict__ be1,
                                float* __restrict__ x1) {
  const int lane = threadIdx.x & 31;
  const int l16 = lane & 15;
  const int hw = lane >> 4;
  const int koff = hw * 2;
  const int n0 = blockIdx.x * 16;

  v8f acc[8] = {};
#pragma unroll 4
  for (int k0 = 0; k0 < C_DIM; k0 += 4) {
    v2f b;
    b.x = attn[(size_t)(k0 + koff) * N_DIM + n0 + l16];
    b.y = attn[(size_t)(k0 + koff + 1) * N_DIM + n0 + l16];
#pragma unroll
    for (int mt = 0; mt < 8; ++mt) {
      v2f a;
      a.x = Wo[(mt * 16 + l16) * C_DIM + k0 + koff];
      a.y = Wo[(mt * 16 + l16) * C_DIM + k0 + koff + 1];
      acc[mt] = wmma_f32(a, b, acc[mt]);
    }
  }
  // Residual + per-voxel LN over the 128 channels.
  float sum = 0.0f, sq = 0.0f;
#pragma unroll
  for (int mt = 0; mt < 8; ++mt) {
#pragma unroll
    for (int r = 0; r < 8; ++r) {
      const int o = mt * 16 + r + 8 * hw;
      const float val = acc[mt][r] + bo[o] + x[(size_t)o * N_DIM + n0 + l16];
      acc[mt][r] = val;
      sum += val;
      sq += val * val;
    }
  }
  sum += __shfl_xor(sum, 16, 32);   // other half-wave holds the other 64 channels
  sq += __shfl_xor(sq, 16, 32);
  const float mean = sum * (1.0f / 128.0f);
  const float var = sq * (1.0f / 128.0f) - mean * mean;
  const float rstd = rsqrtf(var + EPS);
  const int n = n0 + l16;
#pragma unroll
  for (int mt = 0; mt < 8; ++mt) {
#pragma unroll
    for (int r = 0; r < 8; ++r) {
      const int o = mt * 16 + r + 8 * hw;
      x1[(size_t)n * C_DIM + o] = (acc[mt][r] - mean) * rstd * g1[o] + be1[o];
    }
  }
}

// ---------------------------------------------------------------------------
// Kernel 4: FFN (128->2048->128, ReLU) + residual + LayerNorm2 + final store.
// 8 waves per block, 16 voxels per block. FF processed in 4 chunks of 512 so
// the hidden tile (16x512 fp32 = 32 KB) lives in LDS. Output is [C][N].
// ---------------------------------------------------------------------------
__global__ void ffn_ln2_kernel(const float* __restrict__ x1,
                               const float* __restrict__ W1,
                               const float* __restrict__ b1,
                               const float* __restrict__ W2,
                               const float* __restrict__ b2,
                               const float* __restrict__ g2,
                               const float* __restrict__ be2,
                               float* __restrict__ out) {
  __shared__ float hbuf[16 * 512];   // 32 KB hidden chunk
  __shared__ float redS[8][16];
  __shared__ float redQ[8][16];

  const int wave = threadIdx.x >> 5;
  const int lane = threadIdx.x & 31;
  const int l16 = lane & 15;
  const int hw = lane >> 4;
  const int koff = hw * 2;
  const int nb = blockIdx.x;         // 16-voxel block

  const int c0 = wave * 16;          // this wave's output-channel tile
  v8f acc2 = {};

  for (int chunk = 0; chunk < 4; ++chunk) {
    // Phase 1: H[i,f] = relu(sum_c x1[n,c] W1[f,c] + b1[f]), f in this chunk.
#pragma unroll
    for (int t = 0; t < 4; ++t) {
      const int fl = (wave * 4 + t) * 16;        // f offset inside chunk
      const int fg = chunk * 512 + fl;           // global f offset
      v8f hh = {};
#pragma unroll 4
      for (int k0 = 0; k0 < C_DIM; k0 += 4) {
        v2f a, b;
        a.x = x1[(size_t)(nb * 16 + l16) * C_DIM + k0 + koff];
        a.y = x1[(size_t)(nb * 16 + l16) * C_DIM + k0 + koff + 1];
        b.x = W1[(size_t)(fg + l16) * C_DIM + k0 + koff];
        b.y = W1[(size_t)(fg + l16) * C_DIM + k0 + koff + 1];
        hh = wmma_f32(a, b, hh);
      }
      const float bias1 = b1[fg + l16];
#pragma unroll
      for (int r = 0; r < 8; ++r) {
        const int i = r + 8 * hw;
        hbuf[i * 512 + fl + l16] = fmaxf(hh[r] + bias1, 0.0f);
      }
    }
    __syncthreads();
    // Phase 2: out[i,c] += sum_f H[i,f] W2[c,f]   (c in this wave's tile)
#pragma unroll 8
    for (int ks = 0; ks < 128; ++ks) {
      v2f a, b;
      a.x = hbuf[l16 * 512 + ks * 4 + koff];
      a.y = hbuf[l16 * 512 + ks * 4 + koff + 1];
      b.x = W2[(size_t)(c0 + l16) * FF_DIM + chunk * 512 + ks * 4 + koff];
      b.y = W2[(size_t)(c0 + l16) * FF_DIM + chunk * 512 + ks * 4 + koff + 1];
      acc2 = wmma_f32(a, b, acc2);
    }
    __syncthreads();
  }

  // Residual + LN2 across the 128 channels (spread over 8 waves).
  const int c = c0 + l16;
  const float b2c = b2[c];
  const float g2c = g2[c];
  const float be2c = be2[c];
  float vout[8];
#pragma unroll
  for (int r = 0; r < 8; ++r) {
    const int i = r + 8 * hw;
    const int n = nb * 16 + i;
    const float val = acc2[r] + b2c + x1[(size_t)n * C_DIM + c];
    vout[r] = val;
    const float s = redsum16(val);
    const float s2 = redsum16(val * val);
    if (l16 == 0) { redS[wave][i] = s; redQ[wave][i] = s2; }
  }
  __syncthreads();
#pragma unroll
  for (int r = 0; r < 8; ++r) {
    const int i = r + 8 * hw;
    const int n = nb * 16 + i;
    float S = 0.0f, Q = 0.0f;
#pragma unroll
    for (int w2 = 0; w2 < 8; ++w2) { S += redS[w2][i]; Q += redQ[w2][i]; }
    const float mean = S * (1.0f / 128.0f);
    const float var = Q * (1.0f / 128.0f) - mean * mean;
    const float rstd = rsqrtf(var + EPS);
    out[(size_t)c * N_DIM + n] = (vout[r] - mean) * rstd * g2c + be2c;
  }
}

// ---------------------------------------------------------------------------
extern "C" void kernel_launch(void* const* d_in, const int* in_sizes, int n_in,
                              void* d_out, int out_size, void* d_ws, size_t ws_size,
                              hipStream_t stream) {
  const float* x  = (const float*)d_in[0];
  const float* Wq = (const float*)d_in[1];
  const float* bq = (const float*)d_in[2];
  const float* Wk = (const float*)d_in[3];
  const float* bk = (const float*)d_in[4];
  const float* Wv = (const float*)d_in[5];
  const float* bv = (const float*)d_in[6];
  const float* Wo = (const float*)d_in[7];
  const float* bo = (const float*)d_in[8];
  const float* W1 = (const float*)d_in[9];
  const float* b1 = (const float*)d_in[10];
  const float* W2 = (const float*)d_in[11];
  const float* b2 = (const float*)d_in[12];
  const float* g1 = (const float*)d_in[13];
  const float* be1 = (const float*)d_in[14];
  const float* g2 = (const float*)d_in[15];
  const float* be2 = (const float*)d_in[16];

  float* ws = (float*)d_ws;
  const size_t CN = (size_t)C_DIM * N_DIM;     // 524288 floats
  float* q    = ws;
  float* k    = ws + CN;
  float* v    = ws + 2 * CN;
  float* attn = ws + 3 * CN;
  float* x1   = ws + 4 * CN;                   // [N][C]
  // total workspace use: 5 * 2 MB = 10 MB

  qkv_proj_kernel<<<dim3(N_DIM / 16, C_DIM / 16, 3), 32, 0, stream>>>(
      x, Wq, bq, Wk, bk, Wv, bv, q, k, v);
  attention_kernel<<<dim3(N_DIM / 16, 8), 32, 0, stream>>>(q, k, v, attn);
  proj_ln1_kernel<<<N_DIM / 16, 32, 0, stream>>>(attn, Wo, bo, x, g1, be1, x1);
  ffn_ln2_kernel<<<N_DIM / 16, 256, 0, stream>>>(x1, W1, b1, W2, b2, g2, be2,
                                                 (float*)d_out);
}